// TransferNet_29334626632407
// MI455X (gfx1250) — compile-verified
//
#include <hip/hip_runtime.h>
#include <math.h>

#define B_ROWS 4096
#define F_IN   2048
#define F_BOT  256
#define NCLS   31
#define KB     256          // K-block staged in LDS for the bottleneck GEMM
#define LDS_P  260          // padded row stride (floats) to avoid bank conflicts

typedef __attribute__((ext_vector_type(2))) float v2f;
typedef __attribute__((ext_vector_type(8))) float v8f;

// workspace layout (float offsets)
#define WS_SRC   0
#define WS_TAR   (WS_SRC + B_ROWS * F_BOT)      // 1048576
#define WS_S2    (WS_TAR + B_ROWS * F_BOT)      // 2097152
#define WS_T2    (WS_S2 + B_ROWS)               // 2101248
#define WS_DELTA (WS_T2 + B_ROWS)               // 2105344
#define WS_GK    (WS_DELTA + F_BOT)             // 2105600 (+2)

// ---------------------------------------------------------------- init
__global__ __launch_bounds__(256) void k_init(float* __restrict__ out,
                                              float* __restrict__ ws) {
    int t = threadIdx.x;
    if (t == 0) out[0] = 0.f;
    if (t < F_BOT) ws[WS_DELTA + t] = 0.f;
    if (t < 2) ws[WS_GK + t] = 0.f;
}

// ------------------------------------------------- bottleneck GEMM+ReLU
// Block = 8 waves sharing one 16-col N-tile; wave w owns M-tile (mg*8+w).
// W_b K-panel (KB x 16) staged transposed in LDS once per K-block and
// shared by all 8 waves; WMMA = V_WMMA_F32_16X16X4_F32, K-loop 512 total.
__global__ __launch_bounds__(256) void k_bottleneck(const float* __restrict__ src_in,
                                                    const float* __restrict__ tar_in,
                                                    const float* __restrict__ Wb,
                                                    const float* __restrict__ bb,
                                                    float* __restrict__ ws) {
    __shared__ float sW[16 * LDS_P];             // sW[col][k] transposed panel

    const int wave = threadIdx.x >> 5;
    const int lane = threadIdx.x & 31;
    const int blk  = blockIdx.x;                 // 0..1023
    const int mat  = blk >> 9;                   // 0 = source, 1 = target
    const int rem  = blk & 511;                  // 16 n-tiles x 32 m-groups
    const int tn   = rem & 15;
    const int mg   = rem >> 4;

    const float* X = mat ? tar_in : src_in;
    float*       H = ws + (mat ? WS_TAR : WS_SRC);

    const int n0  = tn * 16;
    const int m0  = mg * 128 + wave * 16;
    const int l15 = lane & 15;
    const int lhi = lane >> 4;                   // 0: K=k,k+1  1: K=k+2,k+3

    // staging indices: j = col within panel, i sweeps K rows (coalesced 64B runs)
    const int sj = threadIdx.x & 15;
    const int si = threadIdx.x >> 4;

    const float* arow = X + (size_t)(m0 + l15) * F_IN + lhi * 2;
    const int    bcol = n0 + l15;

    v8f acc = {};
    for (int kb0 = 0; kb0 < F_IN; kb0 += KB) {
        __syncthreads();
#pragma unroll
        for (int r = 0; r < KB / 16; ++r) {
            int i = r * 16 + si;                 // K row within panel
            sW[sj * LDS_P + i] = Wb[(size_t)(kb0 + i) * F_BOT + n0 + sj];
        }
        __syncthreads();

        const float* ap = arow + kb0;
        const float* bp = &sW[l15 * LDS_P + lhi * 2];
#pragma unroll 4
        for (int k = 0; k < KB; k += 4) {
            v2f a = *(const v2f*)(ap + k);       // global b64, contiguous
            v2f b = *(const v2f*)(bp + k);       // LDS b64 from staged panel
            acc = __builtin_amdgcn_wmma_f32_16x16x4_f32(false, a, false, b,
                                                        (short)0, acc, false, false);
        }
    }

    // D layout: col = n0+l15; rows = m0 + lhi*8 + r
    float bias  = bb[bcol];
    int   rbase = m0 + lhi * 8;
#pragma unroll
    for (int r = 0; r < 8; ++r) {
        float v = acc[r] + bias;
        H[(size_t)(rbase + r) * F_BOT + bcol] = v > 0.f ? v : 0.f;
    }
}

// --------------------------------- classifier + softmax margins + CE + s2/t2
__global__ __launch_bounds__(128) void k_classifier(const float* __restrict__ Wc,
                                                    const float* __restrict__ bc,
                                                    const int* __restrict__ labels,
                                                    float* __restrict__ ws,
                                                    float* __restrict__ out) {
    __shared__ float sWc[F_BOT * NCLS];
    __shared__ float sbc[NCLS];
    for (int i = threadIdx.x; i < F_BOT * NCLS; i += blockDim.x) sWc[i] = Wc[i];
    if (threadIdx.x < NCLS) sbc[threadIdx.x] = bc[threadIdx.x];
    __syncthreads();

    int  r      = blockIdx.x * 128 + threadIdx.x;  // 0..8191
    bool is_src = r < B_ROWS;
    int  row    = is_src ? r : r - B_ROWS;
    const float* h = ws + (is_src ? WS_SRC : WS_TAR) + (size_t)row * F_BOT;

    float acc[NCLS];
#pragma unroll
    for (int c = 0; c < NCLS; ++c) acc[c] = sbc[c];

    float ss = 0.f;
    for (int k4 = 0; k4 < F_BOT; k4 += 4) {
        float4 hv = *(const float4*)(h + k4);    // global_load_b128
        ss += hv.x * hv.x + hv.y * hv.y + hv.z * hv.z + hv.w * hv.w;
#pragma unroll
        for (int c = 0; c < NCLS; ++c) {
            acc[c] += hv.x * sWc[(k4 + 0) * NCLS + c];
            acc[c] += hv.y * sWc[(k4 + 1) * NCLS + c];
            acc[c] += hv.z * sWc[(k4 + 2) * NCLS + c];
            acc[c] += hv.w * sWc[(k4 + 3) * NCLS + c];
        }
    }
    ws[(is_src ? WS_S2 : WS_T2) + row] = ss;     // row ||h||^2 for pairwise d2

    float mx = acc[0]; int amax = 0;
#pragma unroll
    for (int c = 1; c < NCLS; ++c)
        if (acc[c] > mx) { mx = acc[c]; amax = c; }

    float conf[NCLS];
    float sum = 0.f;
#pragma unroll
    for (int c = 0; c < NCLS; ++c) { conf[c] = __expf(acc[c] - mx); sum += conf[c]; }
    float inv   = 1.f / sum;
    int   label = is_src ? labels[row] : amax;
    float lc    = conf[label] * inv;
    float best  = 0.f;
#pragma unroll
    for (int c = 0; c < NCLS; ++c) {
        float v = conf[c] * inv;
        if (c != label && v > best) best = v;
    }
    out[4 + r] = lc - best;

    if (is_src) {
        float logp = acc[label] - mx - __logf(sum);
        atomicAdd(out, -logp * (1.f / B_ROWS));  // clf_loss
    }
}

// ------------------------------------------------- column mean delta (MMD)
__global__ __launch_bounds__(256) void k_colmean(float* __restrict__ ws) {
    int t  = threadIdx.x;            // column
    int r0 = blockIdx.x * 32;        // 128 blocks x 32 rows
    const float* S = ws + WS_SRC;
    const float* T = ws + WS_TAR;
    float acc = 0.f;
    for (int i = 0; i < 32; ++i)
        acc += S[(size_t)(r0 + i) * F_BOT + t] - T[(size_t)(r0 + i) * F_BOT + t];
    atomicAdd(&ws[WS_DELTA + t], acc);
}

// ---------------------- pairwise sqdist GEMM fused with exp-kernel reduction
// Block = 8 waves sharing one src tile (tp) staged in LDS; wave w handles
// tar tile tq = qg*8+w. 64 f32 WMMAs per tile over K=256, fused exp+reduce.
__global__ __launch_bounds__(256) void k_pairwise(float* __restrict__ ws) {
    __shared__ float sS[16 * LDS_P];             // src tile [row][k], padded

    const int wave = threadIdx.x >> 5;
    const int lane = threadIdx.x & 31;
    const int blk  = blockIdx.x;                 // 0..8191
    const int tp   = blk >> 5;                   // src tile 0..255
    const int qg   = blk & 31;                   // tar tile group
    const int tq   = qg * 8 + wave;
    const int m0   = tp * 16, n0 = tq * 16;
    const int l15  = lane & 15;
    const int lhi  = lane >> 4;

    // stage src tile: row r, col = threadIdx.x (fully coalesced 1KB rows)
#pragma unroll
    for (int r = 0; r < 16; ++r)
        sS[r * LDS_P + threadIdx.x] = ws[WS_SRC + (size_t)(m0 + r) * F_BOT + threadIdx.x];
    __syncthreads();

    const float* ap = &sS[l15 * LDS_P + lhi * 2];
    const float* T  = ws + WS_TAR + (size_t)(n0 + l15) * F_BOT + lhi * 2;

    v8f acc = {};
#pragma unroll 4
    for (int k = 0; k < F_BOT; k += 4) {
        v2f a = *(const v2f*)(ap + k);           // LDS b64
        v2f b = *(const v2f*)(T + k);            // global b64 (tar row = B col)
        acc = __builtin_amdgcn_wmma_f32_16x16x4_f32(false, a, false, b,
                                                    (short)0, acc, false, false);
    }

    float t2c   = ws[WS_T2 + n0 + l15];
    int   rbase = m0 + lhi * 8;
    float e1 = 0.f, e5 = 0.f;
#pragma unroll
    for (int r = 0; r < 8; ++r) {
        float d2 = ws[WS_S2 + rbase + r] + t2c - 2.f * acc[r];
        d2 = d2 > 0.f ? d2 : 0.f;
        e1 += __expf(-0.5f * d2);
        e5 += __expf(-0.02f * d2);
    }
    for (int off = 16; off > 0; off >>= 1) {     // wave32 shuffle reduction
        e1 += __shfl_down(e1, off, 32);
        e5 += __shfl_down(e5, off, 32);
    }
    if (lane == 0) {
        atomicAdd(&ws[WS_GK + 0], e1);
        atomicAdd(&ws[WS_GK + 1], e5);
    }
}

// ---------------------------------------------------------------- finalize
__global__ __launch_bounds__(256) void k_final(float* __restrict__ ws,
                                               float* __restrict__ out) {
    __shared__ float red[256];
    int t = threadIdx.x;
    float d = ws[WS_DELTA + t] * (1.f / B_ROWS);
    red[t] = d * d;
    __syncthreads();
    for (int s = 128; s > 0; s >>= 1) {
        if (t < s) red[t] += red[t + s];
        __syncthreads();
    }
    if (t == 0) {
        out[1] = red[0];  // MMD
        const float invBB = 1.f / ((float)B_ROWS * (float)B_ROWS);
        out[2] = ws[WS_GK + 0] * invBB;  // gk sigma=1
        out[3] = ws[WS_GK + 1] * invBB;  // gk sigma=5
    }
}

extern "C" void kernel_launch(void* const* d_in, const int* in_sizes, int n_in,
                              void* d_out, int out_size, void* d_ws, size_t ws_size,
                              hipStream_t stream) {
    const float* src = (const float*)d_in[0];
    const float* tar = (const float*)d_in[1];
    const float* Wb  = (const float*)d_in[2];
    const float* bb  = (const float*)d_in[3];
    const float* Wc  = (const float*)d_in[4];
    const float* bc  = (const float*)d_in[5];
    const int*   lab = (const int*)d_in[6];
    float* out = (float*)d_out;
    float* ws  = (float*)d_ws;

    hipLaunchKernelGGL(k_init,       dim3(1),    dim3(256), 0, stream, out, ws);
    hipLaunchKernelGGL(k_bottleneck, dim3(1024), dim3(256), 0, stream, src, tar, Wb, bb, ws);
    hipLaunchKernelGGL(k_classifier, dim3(64),   dim3(128), 0, stream, Wc, bc, lab, ws, out);
    hipLaunchKernelGGL(k_colmean,    dim3(128),  dim3(256), 0, stream, ws);
    hipLaunchKernelGGL(k_pairwise,   dim3(8192), dim3(256), 0, stream, ws);
    hipLaunchKernelGGL(k_final,      dim3(1),    dim3(256), 0, stream, ws, out);
}